// Quantizing_12060268167756
// MI455X (gfx1250) — compile-verified
//
#include <hip/hip_runtime.h>

// ---------------------------------------------------------------------------
// Vector quantization: argmin_q ||x_n - w_q||^2 then gather codebook rows.
// N = 16384 rows, E = 512, Q = 8192.  f32 WMMA (16x16x4) on gfx1250.
//
// Blocking: 256 threads = 8 waves per WG, each wave owns a 16-row strip
// (128 rows/WG -> 128 WGs), looping over all 8192 codebook columns in
// 16-column WMMA tiles staged in LDS.  X block is LDS-resident for the
// whole kernel.  LDS: 264KB (X) + 33KB (W tile) = 297KB <= 320KB.
// W tiles are double-buffered through VGPRs; the workgroup barrier is a
// hand-rolled s_wait_dscnt + s_barrier_signal/wait pair so the in-flight
// global prefetch loads (LOADcnt) stay outstanding across the barrier and
// complete underneath the K loop (hipcc's __syncthreads would force
// s_wait_loadcnt 0 at the barrier and expose the L2 latency).
// ---------------------------------------------------------------------------

typedef __attribute__((ext_vector_type(2))) float v2f;
typedef __attribute__((ext_vector_type(8))) float v8f;

#define NROWS  16384
#define EDIM   512
#define QDIM   8192
#define MBLOCK 128         // rows of x per workgroup (8 waves x 16 rows)
#define QTILE  16          // codebook columns per WMMA tile
#define TPB    256         // 8 waves of 32
#define XPITCH 516         // LDS row pitch in floats; 516 % 64 == 4 -> the 32
                           // lanes' strided B64 reads cover all 64 banks
#define WCHUNK (QTILE * EDIM / 4 / TPB)   // float4 per thread per W tile = 8

// LDS-only workgroup barrier: drain this wave's DS ops, then split-barrier.
// Outstanding global loads (register prefetch) are NOT waited here; their
// consumers get s_wait_loadcnt from normal register dependencies.
__device__ __forceinline__ void wg_barrier_lds() {
    asm volatile("s_wait_dscnt 0x0\n\t"
                 "s_barrier_signal -1\n\t"
                 "s_barrier_wait -1" ::: "memory");
}

// ---------------------------------------------------------------------------
// Kernel 1: wsq[q] = sum_e w[q][e]^2
// ---------------------------------------------------------------------------
__global__ void vq_wsq_kernel(const float* __restrict__ w,
                              float* __restrict__ wsq) {
    int q = blockIdx.x * blockDim.x + threadIdx.x;
    if (q >= QDIM) return;
    const float4* wr = (const float4*)(w + (size_t)q * EDIM);
    float s = 0.0f;
#pragma unroll 8
    for (int i = 0; i < EDIM / 4; ++i) {
        float4 v = wr[i];
        s += v.x * v.x + v.y * v.y + v.z * v.z + v.w * v.w;
    }
    wsq[q] = s;
}

// ---------------------------------------------------------------------------
// Kernel 2: fused GEMM + argmin via V_WMMA_F32_16X16X4_F32.
// score[n][q] = wsq[q] - 2 * dot(x_n, w_q); track running min per row.
// ---------------------------------------------------------------------------
__launch_bounds__(TPB, 1)
__global__ void vq_argmin_kernel(const float* __restrict__ x,
                                 const float* __restrict__ w,
                                 const float* __restrict__ wsq,
                                 int*   __restrict__ best_idx,
                                 float* __restrict__ idx_out) {
    __shared__ float Xs[MBLOCK * XPITCH];   // 128 * 516 * 4B = 264 KB
    __shared__ float Ws[QTILE  * XPITCH];   //  16 * 516 * 4B =  33 KB

    const int tid  = threadIdx.x;
    const int wave = tid >> 5;
    const int lane = tid & 31;
    const int lm   = lane & 15;     // matrix row/col index within lane group
    const int hi   = lane >> 4;     // K sub-offset selector (ISA 16x4 layout)

    const int rowBase = blockIdx.x * MBLOCK;

    // ---- stage 128 rows of x into LDS (float4, padded pitch) ----
    for (int i = tid; i < MBLOCK * (EDIM / 4); i += TPB) {
        int r  = i >> 7;            // EDIM/4 == 128 float4 per row
        int c4 = i & 127;
        float4 v = *(const float4*)(x + (size_t)(rowBase + r) * EDIM + c4 * 4);
        *(float4*)(&Xs[r * XPITCH + c4 * 4]) = v;
    }

    float best[8];
    int   bidx[8];
#pragma unroll
    for (int r = 0; r < 8; ++r) { best[r] = 3.4e38f; bidx[r] = 0; }

    // ---- prime the register prefetch buffer with W tile 0 ----
    float4 pre[WCHUNK];
#pragma unroll
    for (int j = 0; j < WCHUNK; ++j) {
        int i  = tid + j * TPB;
        int r  = i >> 7;
        int c4 = i & 127;
        pre[j] = *(const float4*)(w + (size_t)r * EDIM + c4 * 4);
    }

    // A-matrix layout (16x4 f32): lanes 0-15 hold row M=lane, K = kb + {0,1};
    // lanes 16-31 hold row M=lane-16, K = kb + {2,3}. B (4x16) mirrors it.
    const float* ax = &Xs[(wave * 16 + lm) * XPITCH + 2 * hi];

    for (int qb = 0; qb < QDIM; qb += QTILE) {
        wg_barrier_lds();  // all waves finished reading the previous W tile
                           // (first iteration: also covers the X staging)

        // ---- reg -> LDS: commit the prefetched W tile ----
#pragma unroll
        for (int j = 0; j < WCHUNK; ++j) {
            int i  = tid + j * TPB;
            int r  = i >> 7;
            int c4 = i & 127;
            *(float4*)(&Ws[r * XPITCH + c4 * 4]) = pre[j];
        }

        // ---- issue next tile's global loads; they stay outstanding across
        //      the barrier and complete under the K loop below (wrap keeps
        //      addresses in-bounds on the last tile) ----
        int qn = (qb + QTILE) & (QDIM - 1);
#pragma unroll
        for (int j = 0; j < WCHUNK; ++j) {
            int i  = tid + j * TPB;
            int r  = i >> 7;
            int c4 = i & 127;
            pre[j] = *(const float4*)(w + (size_t)(qn + r) * EDIM + c4 * 4);
        }

        // hoisted: wsq load overlaps the whole K loop
        float wq  = wsq[qb + lm];
        int   qix = qb + lm;

        wg_barrier_lds();  // W tile visible to all waves

        const float* bw = &Ws[lm * XPITCH + 2 * hi];

        // ---- K loop: 128 WMMAs over 4 independent accumulator chains ----
        v8f acc0 = {}, acc1 = {}, acc2 = {}, acc3 = {};
#pragma unroll 2
        for (int k = 0; k < EDIM; k += 16) {
            v2f a0 = *(const v2f*)(ax + k);
            v2f b0 = *(const v2f*)(bw + k);
            acc0 = __builtin_amdgcn_wmma_f32_16x16x4_f32(
                false, a0, false, b0, (short)0, acc0, false, false);
            v2f a1 = *(const v2f*)(ax + k + 4);
            v2f b1 = *(const v2f*)(bw + k + 4);
            acc1 = __builtin_amdgcn_wmma_f32_16x16x4_f32(
                false, a1, false, b1, (short)0, acc1, false, false);
            v2f a2 = *(const v2f*)(ax + k + 8);
            v2f b2 = *(const v2f*)(bw + k + 8);
            acc2 = __builtin_amdgcn_wmma_f32_16x16x4_f32(
                false, a2, false, b2, (short)0, acc2, false, false);
            v2f a3 = *(const v2f*)(ax + k + 12);
            v2f b3 = *(const v2f*)(bw + k + 12);
            acc3 = __builtin_amdgcn_wmma_f32_16x16x4_f32(
                false, a3, false, b3, (short)0, acc3, false, false);
        }

        // ---- epilogue: scores + running argmin (per lane, 8 rows) ----
        v8f acc = acc0 + acc1 + acc2 + acc3;
#pragma unroll
        for (int r = 0; r < 8; ++r) {
            float score = fmaf(-2.0f, acc[r], wq);
            if (score < best[r]) { best[r] = score; bidx[r] = qix; }
        }
    }

    // ---- cross-lane reduction: 16 candidate columns per row group ----
    // C/D layout: VGPR r = row M=r (lanes 0-15) / M=r+8 (lanes 16-31).
#pragma unroll
    for (int r = 0; r < 8; ++r) {
        float v  = best[r];
        int   bi = bidx[r];
#pragma unroll
        for (int m = 8; m >= 1; m >>= 1) {
            float ov = __shfl_xor(v, m, 16);
            int   oi = __shfl_xor(bi, m, 16);
            if (ov < v || (ov == v && oi < bi)) { v = ov; bi = oi; }
        }
        best[r] = v;
        bidx[r] = bi;
    }

    if (lm == 0) {   // lane 0 -> rows 0..7 of strip, lane 16 -> rows 8..15
#pragma unroll
        for (int r = 0; r < 8; ++r) {
            int n = rowBase + wave * 16 + hi * 8 + r;
            best_idx[n] = bidx[r];
            idx_out[n]  = (float)bidx[r];
        }
    }
}

// ---------------------------------------------------------------------------
// Kernel 3: gather codebook rows into output.
// ---------------------------------------------------------------------------
__global__ void vq_gather_kernel(const float* __restrict__ w,
                                 const int* __restrict__ bidx,
                                 float* __restrict__ out) {
    int n = blockIdx.x;
    int q = bidx[n];
    const float4* src = (const float4*)(w + (size_t)q * EDIM);
    float4*       dst = (float4*)(out + (size_t)n * EDIM);
    dst[threadIdx.x] = src[threadIdx.x];
}

// ---------------------------------------------------------------------------
extern "C" void kernel_launch(void* const* d_in, const int* in_sizes, int n_in,
                              void* d_out, int out_size, void* d_ws, size_t ws_size,
                              hipStream_t stream) {
    const float* x = (const float*)d_in[0];   // (8, 2048, 512) f32
    const float* w = (const float*)d_in[1];   // (8192, 512)    f32

    float* out_q   = (float*)d_out;                       // N*E floats
    float* out_idx = out_q + (size_t)NROWS * EDIM;        // N floats (idx)

    float* wsq  = (float*)d_ws;                           // Q floats
    int*   bidx = (int*)((char*)d_ws + QDIM * sizeof(float)); // N ints

    vq_wsq_kernel<<<QDIM / 256, 256, 0, stream>>>(w, wsq);
    vq_argmin_kernel<<<NROWS / MBLOCK, TPB, 0, stream>>>(x, w, wsq, bidx, out_idx);
    vq_gather_kernel<<<NROWS, EDIM / 4, 0, stream>>>(w, bidx, out_q);
}